// RVAEModel_69252052681266
// MI455X (gfx1250) — compile-verified
//
#include <hip/hip_runtime.h>

// VQ-VAE codebook gather + [bs,8,8,C] -> [bs,C,8,8] permute.
// Pure data movement: 64MB stores dominate (weight table is L2-resident,
// ~4MB actual HBM reads) -> ~3us floor at 23.3 TB/s. No WMMA: emulating the
// one-hot matmul would add 34 GFLOP + 64GB of one-hot traffic for nothing.
// Strategy: per-block tile (one sample b, 64 channels), async-copy the 64
// gathered rows *transposed* straight into LDS via the CDNA5 async
// global->LDS path (ASYNCcnt / s_wait_asynccnt), then emit fully-coalesced
// float4 stores (512B contiguous per wave).

#define BS_     256
#define M_      64      // tokens per sample (8x8)
#define C_      1024    // channels
#define TILE_C  64      // channels per block
#define LSTRIDE 65      // odd stride: bank-conflict-free transposed LDS tile

typedef __attribute__((address_space(1))) int as1_int_t; // global
typedef __attribute__((address_space(3))) int as3_int_t; // LDS

__global__ __launch_bounds__(256) void vq_gather_permute_kernel(
    const int* __restrict__ idx,      // [BS, M]
    const float* __restrict__ weight, // [NUM_EMB, C]
    float* __restrict__ out)          // [BS, C, M]
{
    __shared__ float tileT[TILE_C * LSTRIDE]; // tileT[c_local*65 + g]
    __shared__ int   sidx[M_];

    const int tid = threadIdx.x;
    const int b   = blockIdx.x >> 4;          // 16 channel-tiles per sample
    const int c0  = (blockIdx.x & 15) << 6;   // tile base channel

    if (tid < M_) sidx[tid] = idx[b * M_ + tid];
    __syncthreads();

#if __has_builtin(__builtin_amdgcn_global_load_async_to_lds_b32)
    // Phase 1 (async path): gather + transpose during the copy.
    // e = tid + 256k : g = e>>6 (token row), c = e&63 (channel in tile).
    // Lanes span c -> 128B contiguous global reads per instruction; LDS
    // stride 65 -> all 32 async LDS writes hit distinct banks.
#pragma unroll
    for (int k = 0; k < 16; ++k) {
        const int e = tid + 256 * k;
        const int g = e >> 6;
        const int c = e & 63;
        const float* gsrc = weight + (size_t)sidx[g] * C_ + (c0 + c);
        float*       ldst = &tileT[c * LSTRIDE + g];
        __builtin_amdgcn_global_load_async_to_lds_b32(
            (as1_int_t*)gsrc, (as3_int_t*)ldst, /*offset=*/0, /*cpol=*/0);
    }
  #if __has_builtin(__builtin_amdgcn_s_wait_asynccnt)
    __builtin_amdgcn_s_wait_asynccnt(0);
  #else
    asm volatile("s_wait_asynccnt 0" ::: "memory");
  #endif
#else
    // Fallback: float4 gather through VGPRs, scalar transposed LDS stores
    // (addr = 65*c + g, lane deltas 4*65=260 -> 32 distinct banks).
#pragma unroll
    for (int k = 0; k < 4; ++k) {
        const int q  = tid + 256 * k;   // float4 id: row g = q>>4, f4 = q&15
        const int g  = q >> 4;
        const int f4 = q & 15;
        const float4 v = *reinterpret_cast<const float4*>(
            weight + (size_t)sidx[g] * C_ + c0 + 4 * f4);
        const int cl = 4 * f4;
        tileT[(cl + 0) * LSTRIDE + g] = v.x;
        tileT[(cl + 1) * LSTRIDE + g] = v.y;
        tileT[(cl + 2) * LSTRIDE + g] = v.z;
        tileT[(cl + 3) * LSTRIDE + g] = v.w;
    }
#endif
    __syncthreads();

    // Phase 2: block's output region is one contiguous 16KB span:
    // out[b*C*M + (c0+cl)*64 + g]. Each wave stores 512B contiguous.
    // LDS reads: addr = 65*cl + 4*(i4&15) + j -> 32 distinct banks per j.
    float* outBase = out + (size_t)b * (C_ * M_) + (size_t)c0 * M_;
#pragma unroll
    for (int k = 0; k < 4; ++k) {
        const int i4 = tid + 256 * k;       // output float4 id within block
        const int cl = i4 >> 4;             // channel in tile
        const int g4 = (i4 & 15) << 2;      // starting token
        float4 v;
        v.x = tileT[cl * LSTRIDE + g4 + 0];
        v.y = tileT[cl * LSTRIDE + g4 + 1];
        v.z = tileT[cl * LSTRIDE + g4 + 2];
        v.w = tileT[cl * LSTRIDE + g4 + 3];
        *reinterpret_cast<float4*>(outBase + cl * 64 + g4) = v;
    }
}

extern "C" void kernel_launch(void* const* d_in, const int* in_sizes, int n_in,
                              void* d_out, int out_size, void* d_ws, size_t ws_size,
                              hipStream_t stream) {
    (void)in_sizes; (void)n_in; (void)d_ws; (void)ws_size; (void)out_size;
    const int*   idx    = (const int*)d_in[0];     // encoding_indices [256,64]
    const float* weight = (const float*)d_in[1];   // [1024,1024]
    float*       out    = (float*)d_out;           // [256,1024,8,8]

    dim3 grid(BS_ * (C_ / TILE_C));  // 4096 blocks
    dim3 block(256);                 // 8 wave32s
    hipLaunchKernelGGL(vq_gather_permute_kernel, grid, block, 0, stream,
                       idx, weight, out);
}